// PKM_26010321945096
// MI455X (gfx1250) — compile-verified
//
#include <hip/hip_runtime.h>
#include <hip/hip_bf16.h>

#define DIM      512
#define HEADS    8
#define NUM_KEYS 256
#define TOPK     16
#define NTOK     4096        // B*T = 2*2048
#define EPS_LN   1e-5f
#define TPB_TOK  8           // tokens per workgroup in pkm_attend
#define QN_STRIDE (DIM + 4)  // 516: rows 16B-aligned for async b128; banks (4*row+c)%64
                             // distinct over the 16-lane A-fragment reads -> conflict-free

typedef float v2f __attribute__((ext_vector_type(2)));
typedef float v8f __attribute__((ext_vector_type(8)));
typedef int   v4i __attribute__((ext_vector_type(4)));

#if __has_builtin(__builtin_amdgcn_global_load_async_to_lds_b128) && \
    __has_builtin(__builtin_amdgcn_s_wait_asynccnt)
#define HAVE_ASYNC_LDS 1
typedef __attribute__((address_space(1))) v4i gas_v4i;   // global int4
typedef __attribute__((address_space(3))) v4i las_v4i;   // LDS int4
#else
#define HAVE_ASYNC_LDS 0
#endif

__device__ __forceinline__ v8f wmma_f32x4(v2f a, v2f b, v8f c) {
    // D = A(16x4) * B(4x16) + C, fp32 (confirmed: lowers to v_wmma_f32_16x16x4_f32)
    return __builtin_amdgcn_wmma_f32_16x16x4_f32(false, a, false, b, (short)0, c,
                                                 false, false);
}

// ---------------------------------------------------------------------------
// Kernel 1: q = x @ Wq^T   (q[n,f] = sum_e x[n,e] * Wq[f,e])
// One wave32 per 16x16 output tile; fp32 WMMA accumulation over K=512.
// ---------------------------------------------------------------------------
__global__ __launch_bounds__(256)
void pkm_qgemm(const float* __restrict__ x, const float* __restrict__ Wq,
               float* __restrict__ q)
{
    const int lane = threadIdx.x & 31;
    const int wave = threadIdx.x >> 5;
    const int tile = blockIdx.x * 8 + wave;          // 8192 tiles total
    const int tiles_n = DIM / 16;                    // 32
    const int tm = tile / tiles_n;                   // token-tile  [0,256)
    const int tn = tile % tiles_n;                   // feature-tile[0,32)
    const int half = lane >> 4;
    const int l16  = lane & 15;

    // A row m = l16 (x), B col n = l16 (B[k][n] = Wq[fbase+n, k])
    const float* arow = x  + (size_t)(tm * 16 + l16) * DIM;
    const float* bcol = Wq + (size_t)(tn * 16 + l16) * DIM;

    v8f acc = {};
    for (int kb = 0; kb < DIM; kb += 4) {
        v2f a, b;
        a.x = arow[kb + 2 * half + 0];
        a.y = arow[kb + 2 * half + 1];
        b.x = bcol[kb + 2 * half + 0];
        b.y = bcol[kb + 2 * half + 1];
        acc = wmma_f32x4(a, b, acc);
    }

    // C/D layout: VGPR r -> row (r + 8*half), col l16
#pragma unroll
    for (int r = 0; r < 8; ++r)
        q[(size_t)(tm * 16 + r + 8 * half) * DIM + tn * 16 + l16] = acc[r];
}

// ---------------------------------------------------------------------------
// Kernel 2: per-feature mean / rsqrt(var+eps) over all NTOK tokens (axis=0)
// ---------------------------------------------------------------------------
__global__ __launch_bounds__(256)
void pkm_stats(const float* __restrict__ q, float* __restrict__ meanOut,
               float* __restrict__ rstdOut)
{
    const int f = blockIdx.x * 256 + threadIdx.x;   // 2 blocks x 256 = 512 features
    float s = 0.f, s2 = 0.f;
    for (int n = 0; n < NTOK; ++n) {
        float v = q[(size_t)n * DIM + f];
        s += v;
        s2 += v * v;
    }
    float mean = s * (1.0f / NTOK);
    float var  = s2 * (1.0f / NTOK) - mean * mean;  // biased var (matches jnp.var)
    meanOut[f] = mean;
    rstdOut[f] = rsqrtf(var + EPS_LN);
}

// ---------------------------------------------------------------------------
// Kernel 3: 8 tokens per workgroup (256 thr = 8 wave32).
//   async-stage 8 q rows to LDS -> normalize in place
//   -> per bank: WMMA dots (16x32 @ 32x256, rows 8..15 zero-pad)
//   -> wave-parallel top-16/bank -> combine pairs -> top-16/group -> softmax
//   -> gather 128 value rows per token (L2-resident 128MB table) with prefetch.
// ---------------------------------------------------------------------------
__global__ __launch_bounds__(256)
void pkm_attend(const float* __restrict__ q, const float* __restrict__ mean,
                const float* __restrict__ rstd, const float* __restrict__ gamma,
                const float* __restrict__ beta, const float* __restrict__ keys,
                const float* __restrict__ values, float* __restrict__ out)
{
    __shared__ float qnS[16][QN_STRIDE];            // 8 real rows + 8 zero rows
    __shared__ float dotsBank[TPB_TOK][NUM_KEYS];   // 8KB, reused per bank, then
                                                    // aliased as wAttn/wIdx below
    __shared__ float topS[TPB_TOK][16][TOPK];       // 8KB
    __shared__ int   topI[TPB_TOK][16][TOPK];       // 8KB

    float (*wAttn)[HEADS * TOPK] = (float (*)[HEADS * TOPK])dotsBank;
    int   (*wIdx)[HEADS * TOPK]  = (int (*)[HEADS * TOPK])((char*)dotsBank + 4096);

    const int tid  = threadIdx.x;
    const int lane = tid & 31;
    const int wave = tid >> 5;
    const int half = lane >> 4;
    const int l16  = lane & 15;
    const int tok0 = blockIdx.x * TPB_TOK;
    const float NEG_INF = -__builtin_inff();

    // -------- phase 0: stage 8 token rows to LDS, normalize, zero pad rows ----
#if HAVE_ASYNC_LDS
    // Async DMA: 1024 x b128 chunks (8 rows x 128 float4), direct global->LDS.
#pragma unroll
    for (int it = 0; it < (TPB_TOK * DIM / 4) / 256; ++it) {
        int idx  = tid + it * 256;                   // float4-chunk id
        int trow = idx >> 7;                         // 128 chunks per row
        int f4   = idx & 127;
        const float* gsrc = q + (size_t)(tok0 + trow) * DIM + f4 * 4;
        float*       ldst = &qnS[trow][f4 * 4];      // rows 16B-aligned (stride 516)
        __builtin_amdgcn_global_load_async_to_lds_b128(
            (gas_v4i*)gsrc, (las_v4i*)ldst, 0, 0);
    }
    __builtin_amdgcn_s_wait_asynccnt(0);             // own wave's copies done
    __syncthreads();                                 // => all waves' copies done
    for (int i = tid; i < TPB_TOK * DIM; i += 256) {
        int trow = i >> 9;
        int f    = i & (DIM - 1);
        float v  = qnS[trow][f];
        qnS[trow][f]     = (v - mean[f]) * rstd[f] * gamma[f] + beta[f];
        qnS[trow + 8][f] = 0.f;
    }
#else
    for (int i = tid; i < TPB_TOK * DIM; i += 256) {
        int trow = i >> 9;
        int f    = i & (DIM - 1);
        float v  = q[(size_t)(tok0 + trow) * DIM + f];
        qnS[trow][f]     = (v - mean[f]) * rstd[f] * gamma[f] + beta[f];
        qnS[trow + 8][f] = 0.f;
    }
#endif
    __syncthreads();

    // -------- per-bank: WMMA dots + per-token top-16 --------
    for (int bank = 0; bank < 16; ++bank) {
        const int h = bank >> 1, p = bank & 1;
        const int off = p * 256 + h * 32;            // qs[p, h, :] offset

        // A fragments (M=16 tokens x K=32), hoisted across the 2 n-tiles
        v2f afr[8];
#pragma unroll
        for (int s = 0; s < 8; ++s) {
            afr[s].x = qnS[l16][off + 4 * s + 2 * half + 0];
            afr[s].y = qnS[l16][off + 4 * s + 2 * half + 1];
        }
#pragma unroll
        for (int t = 0; t < 2; ++t) {
            const int ntile = wave * 2 + t;          // 16 n-tiles over 8 waves
            // B[k][n] = keys[h, ntile*16 + n, p, k]
            const float* bbase =
                keys + (size_t)((h * NUM_KEYS + ntile * 16 + l16) * 2 + p) * 32;
            v8f acc = {};
#pragma unroll
            for (int s = 0; s < 8; ++s) {
                v2f b;
                b.x = bbase[4 * s + 2 * half + 0];
                b.y = bbase[4 * s + 2 * half + 1];
                acc = wmma_f32x4(afr[s], b, acc);
            }
            if (half == 0) {                         // rows 0..7 = real tokens
#pragma unroll
                for (int r = 0; r < 8; ++r)
                    dotsBank[r][ntile * 16 + l16] = acc[r];
            }
        }
        __syncthreads();

        // top-16 of 256 for token == wave (iterative wave argmax, low-idx ties)
        {
            float v[8];
#pragma unroll
            for (int j = 0; j < 8; ++j) v[j] = dotsBank[wave][lane + 32 * j];
            float myS = NEG_INF;
            int   myI = 0;
            for (int iter = 0; iter < TOPK; ++iter) {
                float bv = v[0];
                int   bi = lane;
#pragma unroll
                for (int j = 1; j < 8; ++j) {
                    int idx = lane + 32 * j;
                    if (v[j] > bv) { bv = v[j]; bi = idx; }
                }
#pragma unroll
                for (int o = 16; o > 0; o >>= 1) {
                    float ov = __shfl_xor(bv, o, 32);
                    int   oi = __shfl_xor(bi, o, 32);
                    if (ov > bv || (ov == bv && oi < bi)) { bv = ov; bi = oi; }
                }
                if ((bi & 31) == lane) v[bi >> 5] = NEG_INF;
                if (iter == lane) { myS = bv; myI = bi; }   // rank capture
            }
            if (lane < TOPK) {
                topS[wave][bank][lane] = myS;
                topI[wave][bank][lane] = myI;
            }
        }
        __syncthreads();
    }

    // -------- combine banks (g, g+8), top-16 of 256, softmax; token == wave ----
    // group g <-> (head hg = g>>1, part p = g&1): pA = g, pB = g + 8
    for (int g = 0; g < 8; ++g) {
        float v[8];
#pragma unroll
        for (int j = 0; j < 8; ++j) {
            int c = lane + 32 * j;                   // c = i*16 + jj
            v[j] = topS[wave][g][c >> 4] + topS[wave][g + 8][c & 15];
        }
        float myS = NEG_INF;
        int   myB = 0;
        for (int iter = 0; iter < TOPK; ++iter) {
            float bv = v[0];
            int   bi = lane;
#pragma unroll
            for (int j = 1; j < 8; ++j) {
                int idx = lane + 32 * j;
                if (v[j] > bv) { bv = v[j]; bi = idx; }
            }
#pragma unroll
            for (int o = 16; o > 0; o >>= 1) {
                float ov = __shfl_xor(bv, o, 32);
                int   oi = __shfl_xor(bi, o, 32);
                if (ov > bv || (ov == bv && oi < bi)) { bv = ov; bi = oi; }
            }
            if ((bi & 31) == lane) v[bi >> 5] = NEG_INF;
            if (iter == lane) { myS = bv; myB = bi; }
        }
        // softmax over the 16 ranks, all in registers (rank 0 = max)
        float s0  = __shfl(myS, 0, 32);
        float e   = expf(myS - s0);
        float sum = e;
#pragma unroll
        for (int o = 8; o > 0; o >>= 1) sum += __shfl_xor(sum, o, 32);
        if (lane < TOPK) {
            int i = myB >> 4, jj = myB & 15;
            wAttn[wave][g * TOPK + lane] = e / sum;
            wIdx[wave][g * TOPK + lane]  =
                topI[wave][g][i] * NUM_KEYS + topI[wave][g + 8][jj];
        }
    }
    __syncthreads();

    // -------- gather: 128 weighted value rows per token --------
    for (int t = 0; t < TPB_TOK; ++t) {
        float acc0 = 0.f, acc1 = 0.f;
        for (int j = 0; j < HEADS * TOPK; ++j) {
            if (j + 4 < HEADS * TOPK)                // cover index->load latency
                __builtin_prefetch(values + (size_t)wIdx[t][j + 4] * DIM + tid, 0, 1);
            float w = wAttn[t][j];
            const float* vr = values + (size_t)wIdx[t][j] * DIM;
            acc0 += w * vr[tid];
            acc1 += w * vr[tid + 256];
        }
        out[(size_t)(tok0 + t) * DIM + tid]       = acc0;
        out[(size_t)(tok0 + t) * DIM + tid + 256] = acc1;
    }
}

// ---------------------------------------------------------------------------
extern "C" void kernel_launch(void* const* d_in, const int* in_sizes, int n_in,
                              void* d_out, int out_size, void* d_ws, size_t ws_size,
                              hipStream_t stream)
{
    const float* x      = (const float*)d_in[0];
    const float* Wq     = (const float*)d_in[1];
    const float* gamma  = (const float*)d_in[2];
    const float* beta   = (const float*)d_in[3];
    const float* keys   = (const float*)d_in[4];
    const float* values = (const float*)d_in[5];
    float* out = (float*)d_out;

    // q (NTOK x DIM fp32) is exactly the output shape: reuse d_out as scratch.
    // pkm_attend reads its 8 q rows into LDS before overwriting them with output.
    float* qbuf    = out;
    float* meanBuf = (float*)d_ws;        // 512 floats
    float* rstdBuf = meanBuf + DIM;       // 512 floats

    const int tiles = (NTOK / 16) * (DIM / 16);       // 8192 wave-tiles
    pkm_qgemm <<<tiles / 8, 256, 0, stream>>>(x, Wq, qbuf);
    pkm_stats <<<DIM / 256, 256, 0, stream>>>(qbuf, meanBuf, rstdBuf);
    pkm_attend<<<NTOK / TPB_TOK, 256, 0, stream>>>(qbuf, meanBuf, rstdBuf,
                                                   gamma, beta, keys, values, out);
}